// acRKNLayer_63952063038110
// MI455X (gfx1250) — compile-verified
//
#include <hip/hip_runtime.h>
#include <hip/hip_bf16.h>
#include <math.h>

// ---------------------------------------------------------------------------
// Problem constants (from the reference): B=256, T=256, lod=64, lsd=128,
// K=15 (pad 16), act=32, H=60 (pad 64), bandwidth=3 -> 7 diagonals.
// Banded flattening: n = f*448 + i*7 + d,  j = i-3+d,  f in 0..3  -> N=1792.
// ---------------------------------------------------------------------------
#define BB    256
#define TT    256
#define LOD   64
#define LSD   128
#define KM    15
#define ACT   32
#define HH    60
#define NBAND 1792          // 4*64*7
#define NTILE 112           // 1792/16
#define TMS_STR 1800        // padded row stride for per-row banded matrices

typedef __attribute__((ext_vector_type(2))) float v2f;
typedef __attribute__((ext_vector_type(8))) float v8f;

__device__ __forceinline__ v8f wmma4(v2f a, v2f b, v8f c) {
  // D(16x16,f32) = A(16x4,f32) * B(4x16,f32) + C
  return __builtin_amdgcn_wmma_f32_16x16x4_f32(
      /*neg_a=*/false, a, /*neg_b=*/false, b,
      /*c_mod=*/(short)0, c, /*reuse_a=*/false, /*reuse_b=*/false);
}

__device__ __forceinline__ v8f zero8() {
  v8f z = {0.f,0.f,0.f,0.f,0.f,0.f,0.f,0.f};
  return z;
}

__device__ __forceinline__ float elup1(float x) {
  return x >= 0.f ? x + 1.f : __expf(x);
}

// ---------------------------------------------------------------------------
// Kernel 1: extract banded representation of tm_basis * band into workspace.
// tmb[k][f*448 + i*7 + d] = tm_basis[f][k][i][i-3+d]  (0 outside band / k>=15)
// k dimension padded to 16 so it can be the K axis of a WMMA GEMM.
// ---------------------------------------------------------------------------
__global__ void prep_tmb_kernel(const float* __restrict__ tm_basis,
                                float* __restrict__ tmb) {
  int idx = blockIdx.x * 256 + threadIdx.x;
  if (idx >= 16 * NBAND) return;
  int k = idx / NBAND, n = idx % NBAND;
  int f = n / 448, r = n % 448, i = r / 7, d = r % 7;
  int j = i - 3 + d;
  float v = 0.f;
  if (k < KM && j >= 0 && j < LOD)
    v = tm_basis[(((size_t)f * KM + k) * LOD + i) * LOD + j];
  tmb[idx] = v;
}

// ---------------------------------------------------------------------------
// Kernel 2: ctrl MLP for all (B*T) rows with f32 WMMA.
//   Y = relu(X @ W1 + b1) @ W2 + b2     X:(R,32) W1:(32,60) W2:(60,128)
// One wave handles a 16-row tile. H padded 60->64, zeros propagate.
// ---------------------------------------------------------------------------
__global__ void __launch_bounds__(256)
ctrl_mlp_kernel(const float* __restrict__ X,  const float* __restrict__ W1,
                const float* __restrict__ B1, const float* __restrict__ W2,
                const float* __restrict__ B2, float* __restrict__ Y) {
  __shared__ float hid[8][16][68];                 // per-wave hidden tile
  const int lane = threadIdx.x & 31;
  const int warp = threadIdx.x >> 5;
  const int row0 = (blockIdx.x * 8 + warp) * 16;
  const int m  = lane & 15;                        // A-frag row
  const int kk = (lane >> 4) * 2;                  // K sub-offset (0 or 2)
  const int nn = lane & 15;                        // B/C column
  const int mb = (lane >> 4) * 8;                  // C row offset

  // ---- stage 1: hidden = relu(X @ W1 + b1), N padded to 64 ----
  for (int nt = 0; nt < 4; ++nt) {
    v8f c = zero8();
    const int col = nt * 16 + nn;
    for (int ks = 0; ks < 8; ++ks) {               // K = 32
      int k0 = 4 * ks + kk;
      v2f a; a.x = X[(size_t)(row0 + m) * ACT + k0];
             a.y = X[(size_t)(row0 + m) * ACT + k0 + 1];
      v2f bf;
      bf.x = (col < HH) ? W1[(size_t)k0       * HH + col] : 0.f;
      bf.y = (col < HH) ? W1[(size_t)(k0 + 1) * HH + col] : 0.f;
      c = wmma4(a, bf, c);
    }
    float bias = (col < HH) ? B1[col] : 0.f;
    for (int r = 0; r < 8; ++r) {
      float h = c[r] + bias;
      hid[warp][r + mb][col] = h > 0.f ? h : 0.f;
    }
  }
  __syncthreads();

  // ---- stage 2: Y = hidden @ W2 + b2  (K = 64 padded, rows 60..63 zero) ----
  for (int nt = 0; nt < 8; ++nt) {
    v8f c = zero8();
    const int col = nt * 16 + nn;
    for (int ks = 0; ks < 16; ++ks) {
      int k0 = 4 * ks + kk;
      v2f a; a.x = hid[warp][m][k0]; a.y = hid[warp][m][k0 + 1];
      v2f bf;
      bf.x = (k0     < HH) ? W2[(size_t)k0       * LSD + col] : 0.f;
      bf.y = (k0 + 1 < HH) ? W2[(size_t)(k0 + 1) * LSD + col] : 0.f;
      c = wmma4(a, bf, c);
    }
    float bias = B2[col];
    for (int r = 0; r < 8; ++r)
      Y[(size_t)(row0 + r + mb) * LSD + col] = c[r] + bias;
  }
}

// ---------------------------------------------------------------------------
// Kernel 3: the sequential scan. One block = 16 batch rows * 64 lod lanes
// = 1024 threads (32 waves). Carry lives in LDS (~141 KB total).
// The banded-basis B fragments (loop-invariant over t) are preloaded into
// per-wave REGISTERS (<= 4 tiles * 4 ksteps * 2 dwords = 32 VGPRs), so the
// dominant phase-C GEMM runs pure WMMA on register operands.
// Per step:
//   A) posterior update (elementwise, own lanes only)
//   B) logits = post_mean(16x128) @ coeff_w(128x16pad)  -- 32 WMMAs, wave 0
//   C) tms    = coeff(16x16) @ tmb(16x1792)             -- 448 WMMAs, all waves
//   D) banded predict (7-diagonal matvec + covariance), VALU
// ---------------------------------------------------------------------------
__global__ void __launch_bounds__(1024)
rkn_scan_kernel(const float* __restrict__ latent_obs,
                const float* __restrict__ obs_vars,
                const unsigned char* __restrict__ obs_valid,
                const float* __restrict__ initial_mean,
                const float* __restrict__ initial_cov,
                const float* __restrict__ coeff_w,
                const float* __restrict__ coeff_b,
                const float* __restrict__ tcov_u,
                const float* __restrict__ tcov_l,
                const float* __restrict__ tmb,
                const float* __restrict__ ctrl,
                float* __restrict__ o_pm,  float* __restrict__ o_pcu,
                float* __restrict__ o_pcl, float* __restrict__ o_pcs,
                float* __restrict__ o_npm, float* __restrict__ o_ncu,
                float* __restrict__ o_ncl, float* __restrict__ o_ncs) {
  extern __shared__ float smem[];
  float* s_tms = smem;                    // 16 * 1800  (per-row banded tms)
  float* s_pm  = s_tms + 16 * TMS_STR;    // 16 * 128
  float* s_cu  = s_pm  + 16 * LSD;        // 16 * 64
  float* s_cl  = s_cu  + 16 * LOD;
  float* s_cs  = s_cl  + 16 * LOD;
  float* s_cw  = s_cs  + 16 * LOD;        // 128 * 16 (coeff_w, N padded)
  float* s_log = s_cw  + LSD * 16;        // 16 * 16 logits
  float* s_cf  = s_log + 256;             // 16 * 16 softmax coeffs (col 15 = 0)
  float* s_cb  = s_cf  + 256;             // 16

  const int tid  = threadIdx.x;
  const int row  = tid >> 6;              // 0..15 batch row in tile
  const int i    = tid & 63;              // 0..63 lod index
  const int b    = blockIdx.x * 16 + row;
  const int lane = tid & 31;
  const int wv   = tid >> 5;              // wave id 0..31
  const int m  = lane & 15;
  const int kk = (lane >> 4) * 2;
  const int nn = lane & 15;
  const int mb = (lane >> 4) * 8;

  // stage coeff_w into LDS (used by wave 0 every step)
  for (int idx = tid; idx < LSD * 16; idx += 1024) {
    int cdim = idx >> 4, k = idx & 15;
    s_cw[idx] = (k < KM) ? coeff_w[cdim * KM + k] : 0.f;
  }
  if (tid < 16) s_cb[tid] = (tid < KM) ? coeff_b[tid] : 0.f;

  // preload banded-basis B fragments for this wave's N-tiles (t-invariant)
  v2f btile[4][4];
#pragma unroll
  for (int s = 0; s < 4; ++s) {
    const int nt = wv + 32 * s;           // wave-uniform
#pragma unroll
    for (int ks = 0; ks < 4; ++ks) {
      if (nt < NTILE) {
        int k0 = 4 * ks + kk;
        btile[s][ks].x = tmb[(size_t)k0       * NBAND + nt * 16 + nn];
        btile[s][ks].y = tmb[(size_t)(k0 + 1) * NBAND + nt * 16 + nn];
      } else {
        btile[s][ks].x = 0.f; btile[s][ks].y = 0.f;
      }
    }
  }

  // initialize carry
  s_pm[row * LSD + i]       = initial_mean[(size_t)b * LSD + i];
  s_pm[row * LSD + LOD + i] = initial_mean[(size_t)b * LSD + LOD + i];
  s_cu[row * LOD + i] = initial_cov[(size_t)0 * BB * LOD + (size_t)b * LOD + i];
  s_cl[row * LOD + i] = initial_cov[(size_t)1 * BB * LOD + (size_t)b * LOD + i];
  s_cs[row * LOD + i] = initial_cov[(size_t)2 * BB * LOD + (size_t)b * LOD + i];
  const float tcu_i = elup1(tcov_u[i]);
  const float tcl_i = elup1(tcov_l[i]);
  __syncthreads();

  for (int t = 0; t < TT; ++t) {
    const size_t bt = (size_t)b * TT + t;

    // ---- A: posterior update -------------------------------------------
    float obs = latent_obs[bt * LOD + i];
    float ov  = obs_vars[bt * LOD + i];
    bool valid = obs_valid[bt] != 0;
    float cu = s_cu[row * LOD + i], cl = s_cl[row * LOD + i], cs = s_cs[row * LOD + i];
    float pmu = s_pm[row * LSD + i], pml = s_pm[row * LSD + LOD + i];
    float denom = cu + ov;
    float qu = cu / denom, ql = cs / denom;
    float res = obs - pmu;
    float covf = 1.f - qu;
    float pmu2 = valid ? pmu + qu * res : pmu;
    float pml2 = valid ? pml + ql * res : pml;
    float pcu2 = valid ? covf * cu     : cu;
    float pcl2 = valid ? cl - ql * cs  : cl;
    float pcs2 = valid ? covf * cs     : cs;
    s_pm[row * LSD + i] = pmu2; s_pm[row * LSD + LOD + i] = pml2;
    s_cu[row * LOD + i] = pcu2; s_cl[row * LOD + i] = pcl2; s_cs[row * LOD + i] = pcs2;
    o_pm[bt * LSD + i] = pmu2; o_pm[bt * LSD + LOD + i] = pml2;
    o_pcu[bt * LOD + i] = pcu2; o_pcl[bt * LOD + i] = pcl2; o_pcs[bt * LOD + i] = pcs2;
    __syncthreads();

    // ---- B: logits = post_mean @ coeff_w (wave 0, full EXEC) ------------
    if (wv == 0) {
      v8f c = zero8();
      for (int ks = 0; ks < 32; ++ks) {          // K = 128
        int k0 = 4 * ks + kk;
        v2f a; a.x = s_pm[m * LSD + k0]; a.y = s_pm[m * LSD + k0 + 1];
        v2f bf; bf.x = s_cw[k0 * 16 + nn]; bf.y = s_cw[(k0 + 1) * 16 + nn];
        c = wmma4(a, bf, c);
      }
      for (int r = 0; r < 8; ++r)
        s_log[(r + mb) * 16 + nn] = c[r] + s_cb[nn];
    }
    __syncthreads();

    // ---- softmax over K=15 (one thread per batch row) -------------------
    if (tid < 16) {
      float mx = -1e30f;
      for (int k = 0; k < KM; ++k) mx = fmaxf(mx, s_log[tid * 16 + k]);
      float e[KM]; float sum = 0.f;
      for (int k = 0; k < KM; ++k) { e[k] = __expf(s_log[tid * 16 + k] - mx); sum += e[k]; }
      float inv = 1.f / sum;
      for (int k = 0; k < KM; ++k) s_cf[tid * 16 + k] = e[k] * inv;
      s_cf[tid * 16 + KM] = 0.f;                 // K pad
    }
    __syncthreads();

    // ---- C: banded mixture tms = coeff @ tmb (all 32 waves) -------------
    // A fragments: 8 LDS reads per lane per step; B fragments: registers.
    v2f afr[4];
#pragma unroll
    for (int ks = 0; ks < 4; ++ks) {
      afr[ks].x = s_cf[m * 16 + 4 * ks + kk];
      afr[ks].y = s_cf[m * 16 + 4 * ks + kk + 1];
    }
#pragma unroll
    for (int s = 0; s < 4; ++s) {
      const int nt = wv + 32 * s;                // wave-uniform predicate
      if (nt < NTILE) {
        v8f c = zero8();
#pragma unroll
        for (int ks = 0; ks < 4; ++ks)           // K = 16 (padded)
          c = wmma4(afr[ks], btile[s][ks], c);
        for (int r = 0; r < 8; ++r)
          s_tms[(r + mb) * TMS_STR + nt * 16 + nn] = c[r];
      }
    }
    __syncthreads();

    // ---- D: banded predict (7 diagonals) --------------------------------
    float t11[7], t12[7], t21[7], t22[7];
    const float* tr = &s_tms[row * TMS_STR + i * 7];
    for (int d = 0; d < 7; ++d) {
      t11[d] = tr[d];        t12[d] = tr[448 + d];
      t21[d] = tr[896 + d];  t22[d] = tr[1344 + d];
    }
    t11[3] += 1.f; t22[3] += 1.f;                // + eye (j == i diagonal)
    float nmu = 0.f, nml = 0.f, ncu = 0.f, ncl = 0.f, ncs = 0.f;
    for (int d = 0; d < 7; ++d) {
      int j = i - 3 + d;
      int jc = j < 0 ? 0 : (j > 63 ? 63 : j);    // OOB coeffs are 0 anyway
      float mu = s_pm[row * LSD + jc], ml = s_pm[row * LSD + LOD + jc];
      float du = s_cu[row * LOD + jc], dl = s_cl[row * LOD + jc], dsv = s_cs[row * LOD + jc];
      float a11 = t11[d], a12 = t12[d], a21 = t21[d], a22 = t22[d];
      nmu += a11 * mu + a12 * ml;
      nml += a21 * mu + a22 * ml;
      ncu += a11 * a11 * du + 2.f * a11 * a12 * dsv + a12 * a12 * dl;
      ncl += a21 * a21 * du + 2.f * a21 * a22 * dsv + a22 * a22 * dl;
      ncs += a21 * a11 * du + (a22 * a11 + a21 * a12) * dsv + a22 * a12 * dl;
    }
    ncu += tcu_i; ncl += tcl_i;
    float npmu = nmu + ctrl[bt * LSD + i];
    float npml = nml + ctrl[bt * LSD + LOD + i];
    o_npm[bt * LSD + i] = npmu; o_npm[bt * LSD + LOD + i] = npml;
    o_ncu[bt * LOD + i] = ncu; o_ncl[bt * LOD + i] = ncl; o_ncs[bt * LOD + i] = ncs;
    __syncthreads();                              // all reads done before carry write
    s_pm[row * LSD + i] = npmu; s_pm[row * LSD + LOD + i] = npml;
    s_cu[row * LOD + i] = ncu; s_cl[row * LOD + i] = ncl; s_cs[row * LOD + i] = ncs;
    __syncthreads();
  }
}

// ---------------------------------------------------------------------------
extern "C" void kernel_launch(void* const* d_in, const int* in_sizes, int n_in,
                              void* d_out, int out_size, void* d_ws, size_t ws_size,
                              hipStream_t stream) {
  const float* latent_obs   = (const float*)d_in[0];
  const float* obs_vars     = (const float*)d_in[1];
  const float* actions      = (const float*)d_in[2];
  const float* initial_mean = (const float*)d_in[3];
  const float* initial_cov  = (const float*)d_in[4];
  const float* coeff_w      = (const float*)d_in[5];
  const float* coeff_b      = (const float*)d_in[6];
  const float* ctrl_w1      = (const float*)d_in[7];
  const float* ctrl_b1      = (const float*)d_in[8];
  const float* ctrl_w2      = (const float*)d_in[9];
  const float* ctrl_b2      = (const float*)d_in[10];
  const float* tm_basis     = (const float*)d_in[11];
  const float* tcov_u       = (const float*)d_in[12];
  const float* tcov_l       = (const float*)d_in[13];
  const unsigned char* obs_valid = (const unsigned char*)d_in[14];

  const size_t R = (size_t)BB * TT;
  float* out   = (float*)d_out;
  float* o_pm  = out;
  float* o_pcu = o_pm  + R * LSD;
  float* o_pcl = o_pcu + R * LOD;
  float* o_pcs = o_pcl + R * LOD;
  float* o_npm = o_pcs + R * LOD;
  float* o_ncu = o_npm + R * LSD;
  float* o_ncl = o_ncu + R * LOD;
  float* o_ncs = o_ncl + R * LOD;

  float* ctrl = (float*)d_ws;          // R*128 floats (33.5 MB)
  float* tmb  = ctrl + R * LSD;        // 16*1792 floats

  prep_tmb_kernel<<<(16 * NBAND + 255) / 256, 256, 0, stream>>>(tm_basis, tmb);
  ctrl_mlp_kernel<<<(int)(R / 16 / 8), 256, 0, stream>>>(
      actions, ctrl_w1, ctrl_b1, ctrl_w2, ctrl_b2, ctrl);

  const size_t smem =
      (size_t)(16 * TMS_STR + 16 * LSD + 3 * 16 * LOD +
               LSD * 16 + 256 + 256 + 16) * sizeof(float);   // ~141 KB < 320 KB
  hipFuncSetAttribute((const void*)rkn_scan_kernel,
                      hipFuncAttributeMaxDynamicSharedMemorySize, (int)smem);
  rkn_scan_kernel<<<BB / 16, 1024, smem, stream>>>(
      latent_obs, obs_vars, obs_valid, initial_mean, initial_cov,
      coeff_w, coeff_b, tcov_u, tcov_l, tmb, ctrl,
      o_pm, o_pcu, o_pcl, o_pcs, o_npm, o_ncu, o_ncl, o_ncs);
}